// TaylorLayer_55551107006566
// MI455X (gfx1250) — compile-verified
//
#include <hip/hip_runtime.h>

typedef float v2f __attribute__((ext_vector_type(2)));
typedef float v8f __attribute__((ext_vector_type(8)));

#define T_EXP 164   // 8 + 36 + 120 monomials (degree 1..3 of 8 features)
#define NW_PAD 16   // pad 8 outputs to 16 for the WMMA N dimension

// One wave = one 16-row batch tile. D[16x16] = A[16x164] * B[164x16] via 41
// chained V_WMMA_F32_16X16X4_F32. A (the Taylor expansion) is synthesized in
// registers; B (= W^T zero-padded to 16 rows) is staged in LDS.
__global__ __launch_bounds__(256) void taylor_wmma_kernel(
    const float* __restrict__ x, const float* __restrict__ W,
    const float* __restrict__ bias, float* __restrict__ out, int nbatch) {
  __shared__ float Wlds[NW_PAD * T_EXP];

  // Cooperative fill of padded W: row n, col t -> Wlds[n*164 + t]; rows 8..15 = 0
  for (int t = threadIdx.x; t < NW_PAD * T_EXP; t += blockDim.x) {
    int nrow = t / T_EXP;
    int col = t - nrow * T_EXP;
    Wlds[t] = (nrow < 8) ? W[nrow * T_EXP + col] : 0.0f;
  }
  __syncthreads();

  const int lane = threadIdx.x & 31;
  const int wave = threadIdx.x >> 5;
  const int m = lane & 15;   // batch row within tile (A) / output index N (B,D)
  const int hi = lane >> 4;  // K-half selector: lanes 16-31 carry K+2,K+3
  const int batch_base = blockIdx.x * 128 + wave * 16;
  if (batch_base >= nbatch) return;  // wave-uniform guard, EXEC stays all-ones

  // Load this lane's batch row (8 contiguous floats = 2 x b128).
  const float4* xr = (const float4*)(x + (size_t)(batch_base + m) * 8);
  float4 xa = xr[0], xb = xr[1];
  float xv[8] = {xa.x, xa.y, xa.z, xa.w, xb.x, xb.y, xb.z, xb.w};

  // Degree-2 monomials d2[(j,i)] = x_j * x_i, i >= j (36 values, fully in VGPRs)
  float d2[36];
  {
    int c = 0;
#pragma unroll
    for (int j = 0; j < 8; ++j)
#pragma unroll
      for (int i = j; i < 8; ++i) d2[c++] = xv[j] * xv[i];
  }

  // B-fragment source row: lane's output column n = m, K-half offset 2*hi.
  const float* wrow = &Wlds[m * T_EXP + 2 * hi];

  v8f acc = {};  // 16x16 f32 accumulator tile, zero-initialized

#pragma unroll
  for (int kc = 0; kc < 41; ++kc) {
    // Materialize the 4 expansion terms of this K-chunk (indices are
    // compile-time constants after full unroll -> pure register code).
    float v[4];
#pragma unroll
    for (int q = 0; q < 4; ++q) {
      const int t = kc * 4 + q;
      if (t < 8) {
        v[q] = xv[t];                 // degree 1
      } else if (t < 44) {
        v[q] = d2[t - 8];             // degree 2
      } else {                        // degree 3: x_j * d2[mm]
        const int u = t - 44;
        int j, mm;
        if (u < 36)      { j = 0; mm = u;      }
        else if (u < 64) { j = 1; mm = u - 28; }
        else if (u < 85) { j = 2; mm = u - 49; }
        else if (u < 100){ j = 3; mm = u - 64; }
        else if (u < 110){ j = 4; mm = u - 74; }
        else if (u < 116){ j = 5; mm = u - 80; }
        else if (u < 119){ j = 6; mm = u - 83; }
        else             { j = 7; mm = 35;     }
        v[q] = xv[j] * d2[mm];
      }
    }
    // A fragment (16x4 f32): VGPR0 = K (lo) / K+2 (hi), VGPR1 = K+1 / K+3
    v2f av;
    av.x = hi ? v[2] : v[0];
    av.y = hi ? v[3] : v[1];
    // B fragment (4x16 f32): VGPR0 = K-row 0/2, VGPR1 = K-row 1/3 -> ds_load_b64
    v2f bv = *(const v2f*)(wrow + 4 * kc);

    acc = __builtin_amdgcn_wmma_f32_16x16x4_f32(
        /*neg_a=*/false, av, /*neg_b=*/false, bv,
        /*c_mod=*/(short)0, acc, /*reuse_a=*/false, /*reuse_b=*/false);
  }

  // Epilogue: D VGPR v holds row M = v + 8*hi, column N = m. Valid N < 8.
  if (m < 8) {
    float bn = bias[m];
#pragma unroll
    for (int v = 0; v < 8; ++v) {
      int row = batch_base + v + 8 * hi;
      float val = acc[v] + bn;
      val = val > 0.0f ? val : 0.0f;  // ReLU
      out[(size_t)row * 8 + m] = val;
    }
  }
}

extern "C" void kernel_launch(void* const* d_in, const int* in_sizes, int n_in,
                              void* d_out, int out_size, void* d_ws, size_t ws_size,
                              hipStream_t stream) {
  const float* x = (const float*)d_in[0];   // [B, 8, 1] fp32
  const float* W = (const float*)d_in[1];   // [8, 164]  fp32
  const float* b = (const float*)d_in[2];   // [8, 1]    fp32
  float* out = (float*)d_out;               // [B, 8, 1] fp32

  int nbatch = in_sizes[0] / 8;             // 262144
  int blocks = (nbatch + 127) / 128;        // 128 batch rows per 256-thread block
  hipLaunchKernelGGL(taylor_wmma_kernel, dim3(blocks), dim3(256), 0, stream,
                     x, W, b, out, nbatch);
}